// ComplexBatchNorm2D_bmm_21131239097044
// MI455X (gfx1250) — compile-verified
//
#include <hip/hip_runtime.h>

typedef float v2f __attribute__((ext_vector_type(2)));
typedef float v4f __attribute__((ext_vector_type(4)));
typedef float v8f __attribute__((ext_vector_type(8)));

#define C_CH   256
#define HW     4096
#define NBATCH 32
#define NCHUNK (NBATCH * C_CH)   // 8192 chunks of 4096 elements
#define NTOT   (NBATCH * HW)     // 131072 samples per channel
#define EPSV   1e-5f

// d_ws layout (floats): [0,2048)=stats (256ch x 8), [2048,4096)=params (256ch x 8)

__global__ __launch_bounds__(256) void cbn_zero_stats(float* __restrict__ stats) {
    stats[blockIdx.x * 256 + threadIdx.x] = 0.0f;   // grid=8 -> 2048 floats
}

__global__ __launch_bounds__(256) void cbn_stats_kernel(const float* __restrict__ xr,
                                                        const float* __restrict__ xi,
                                                        float* __restrict__ stats) {
    // one block per contiguous (b, c) chunk of 4096 floats in each of xr, xi
    __shared__ float red[8][32 * 9];   // per-wave 32-row x (5 used) col slab, stride 9
    const int tid  = threadIdx.x;
    const int lane = tid & 31;
    const int wave = tid >> 5;
    const int blk  = blockIdx.x;                 // = b*C + c
    const int chan = blk & (C_CH - 1);

    const v4f* xr4 = (const v4f*)xr + (size_t)blk * (HW / 4);
    const v4f* xi4 = (const v4f*)xi + (size_t)blk * (HW / 4);

    float sr = 0.f, si = 0.f, srr = 0.f, sii = 0.f, sri = 0.f;
#pragma unroll
    for (int k = 0; k < 4; ++k) {
        v4f r = xr4[k * 256 + tid];
        v4f i = xi4[k * 256 + tid];
        sr  += r.x + r.y + r.z + r.w;
        si  += i.x + i.y + i.z + i.w;
        srr += r.x * r.x + r.y * r.y + r.z * r.z + r.w * r.w;
        sii += i.x * i.x + i.y * i.y + i.z * i.z + i.w * i.w;
        sri += r.x * i.x + r.y * i.y + r.z * i.z + r.w * i.w;
    }

    float* slab = red[wave];
    slab[lane * 9 + 0] = sr;
    slab[lane * 9 + 1] = si;
    slab[lane * 9 + 2] = srr;
    slab[lane * 9 + 3] = sii;
    slab[lane * 9 + 4] = sri;
    __syncthreads();

    // WMMA column-sum reduction: D = ones(16x4) x B(4x16) + C, chained over 8
    // row-chunks of the wave's 32x5 slab. With A == all-ones, every D row is the
    // column sum regardless of the hardware's K-slot assignment inside a column.
    v2f a; a[0] = 1.0f; a[1] = 1.0f;
    v8f acc = {};
    const int col   = lane & 15;
    const int cc    = (col < 5) ? col : 0;       // cols 5..15 unused -> alias col 0
    const int rbase = (lane >= 16) ? 2 : 0;
#pragma unroll
    for (int c4 = 0; c4 < 8; ++c4) {
        const int r0 = c4 * 4 + rbase;
        v2f b;
        b[0] = slab[r0 * 9 + cc];
        b[1] = slab[(r0 + 1) * 9 + cc];
        acc = __builtin_amdgcn_wmma_f32_16x16x4_f32(false, a, false, b,
                                                    (short)0, acc, false, false);
    }
    // acc[0], lane j (<16) holds column-sum j  (D[0,j], replicated across rows)
    if (lane < 5) {
        atomicAdd(&stats[chan * 8 + lane], acc[0]);
    }
}

__global__ __launch_bounds__(256) void cbn_params_kernel(const float* __restrict__ stats,
                                                         const float* __restrict__ gamma,
                                                         const float* __restrict__ beta,
                                                         float* __restrict__ params) {
    const int c = threadIdx.x;   // 256 channels, 1 block
    const float sr  = stats[c * 8 + 0];
    const float si  = stats[c * 8 + 1];
    const float srr = stats[c * 8 + 2];
    const float sii = stats[c * 8 + 3];
    const float sri = stats[c * 8 + 4];

    const float invN = 1.0f / (float)NTOT;
    const float mr = sr * invN, mi = si * invN;
    // reference adds EPS twice: cov + EPS*I, then M = cov + EPS*I
    const float a = srr * invN - mr * mr + 2.0f * EPSV;
    const float d = sii * invN - mi * mi + 2.0f * EPSV;
    const float b = sri * invN - mr * mi;

    const float s   = sqrtf(a * d - b * b);
    const float t   = sqrtf(a + d + 2.0f * s);
    const float inv = 1.0f / (s * t);
    const float W00 = (d + s) * inv, W01 = -b * inv;
    const float W10 = -b * inv,      W11 = (a + s) * inv;

    const float g00 = gamma[c * 4 + 0], g01 = gamma[c * 4 + 1];
    const float g10 = gamma[c * 4 + 2], g11 = gamma[c * 4 + 3];

    // out[n,e] = sum_f x[n,f] * A[e,f] + (beta[e] - sum_f mean[f]*A[e,f]),  A = gamma @ W
    const float A00 = g00 * W00 + g01 * W10;
    const float A01 = g00 * W01 + g01 * W11;
    const float A10 = g10 * W00 + g11 * W10;
    const float A11 = g10 * W01 + g11 * W11;
    const float c0  = beta[c * 2 + 0] - (mr * A00 + mi * A01);
    const float c1  = beta[c * 2 + 1] - (mr * A10 + mi * A11);

    params[c * 8 + 0] = A00;
    params[c * 8 + 1] = A01;
    params[c * 8 + 2] = A10;
    params[c * 8 + 3] = A11;
    params[c * 8 + 4] = c0;
    params[c * 8 + 5] = c1;
}

__global__ __launch_bounds__(256) void cbn_apply_kernel(const float* __restrict__ xr,
                                                        const float* __restrict__ xi,
                                                        const float* __restrict__ params,
                                                        float* __restrict__ out) {
    const int tid  = threadIdx.x;
    // Reverse block->chunk mapping (LIFO reuse): the stats pass streamed chunks
    // 0..8191, so the 192 MB L2 holds the TAIL of the input when this kernel
    // starts. Reading the tail first converts up to ~192 MB of HBM reads into
    // L2 hits instead of missing everything (streaming wrap-around).
    const int blk  = (NCHUNK - 1) - blockIdx.x;  // = b*C + c
    const int chan = blk & (C_CH - 1);

    const float A00 = params[chan * 8 + 0];
    const float A01 = params[chan * 8 + 1];
    const float A10 = params[chan * 8 + 2];
    const float A11 = params[chan * 8 + 3];
    const float c0  = params[chan * 8 + 4];
    const float c1  = params[chan * 8 + 5];

    const v4f* xr4  = (const v4f*)xr + (size_t)blk * (HW / 4);
    const v4f* xi4  = (const v4f*)xi + (size_t)blk * (HW / 4);
    v4f*       out4 = (v4f*)out + (size_t)blk * (HW / 2);

#pragma unroll
    for (int k = 0; k < 4; ++k) {
        const int idx = k * 256 + tid;
        v4f r = __builtin_nontemporal_load(&xr4[idx]);
        v4f i = __builtin_nontemporal_load(&xi4[idx]);
        v4f o0, o1;
        o0.x = A00 * r.x + A01 * i.x + c0;  o0.y = A10 * r.x + A11 * i.x + c1;
        o0.z = A00 * r.y + A01 * i.y + c0;  o0.w = A10 * r.y + A11 * i.y + c1;
        o1.x = A00 * r.z + A01 * i.z + c0;  o1.y = A10 * r.z + A11 * i.z + c1;
        o1.z = A00 * r.w + A01 * i.w + c0;  o1.w = A10 * r.w + A11 * i.w + c1;
        __builtin_nontemporal_store(o0, &out4[2 * idx]);
        __builtin_nontemporal_store(o1, &out4[2 * idx + 1]);
    }
}

extern "C" void kernel_launch(void* const* d_in, const int* in_sizes, int n_in,
                              void* d_out, int out_size, void* d_ws, size_t ws_size,
                              hipStream_t stream) {
    const float* xr    = (const float*)d_in[0];
    const float* xi    = (const float*)d_in[1];
    const float* gamma = (const float*)d_in[2];
    const float* beta  = (const float*)d_in[3];
    float*       out   = (float*)d_out;
    float*       ws    = (float*)d_ws;

    float* stats  = ws;            // 2048 floats
    float* params = ws + 2048;     // 2048 floats

    cbn_zero_stats<<<8, 256, 0, stream>>>(stats);
    cbn_stats_kernel<<<NCHUNK, 256, 0, stream>>>(xr, xi, stats);
    cbn_params_kernel<<<1, 256, 0, stream>>>(stats, gamma, beta, params);
    cbn_apply_kernel<<<NCHUNK, 256, 0, stream>>>(xr, xi, params, out);
}